// ChamferLoss_17781164606273
// MI455X (gfx1250) — compile-verified
//
#include <hip/hip_runtime.h>

typedef float v2f __attribute__((ext_vector_type(2)));
typedef float v8f __attribute__((ext_vector_type(8)));

#define WAVES_PER_BLOCK 4
#define BLOCK_THREADS   128
#define MT              2      // m-tiles (16 query points each) per wave

// Raw v_min_num_f32 / v_min3_num_f32: skip fminf's canonicalize (inputs are
// squared distances, never SNaN).
__device__ __forceinline__ float fmin_raw(float a, float b) {
    float r;
    asm("v_min_num_f32 %0, %1, %2" : "=v"(r) : "v"(a), "v"(b));
    return r;
}
__device__ __forceinline__ float fmin3_raw(float a, float b, float c) {
    float r;
    asm("v_min3_num_f32 %0, %1, %2, %3" : "=v"(r) : "v"(a), "v"(b), "v"(c));
    return r;
}

__global__ void chamfer_zero_out(float* out) { out[0] = 0.0f; }

// Pack ref points into the exact WMMA B-matrix lane layout (4x16, K-major):
//   lanes 0-15 : VGPR0 = K0 = -2*rx,  VGPR1 = K1 = -2*ry
//   lanes 16-31: VGPR0 = K2 = -2*rz,  VGPR1 = K3 = yy = ||r||^2
// With A's K3 column set to 1.0, D = A*B + 0 = yy_n - 2 q.r directly.
// Layout: [(b*ntiles + nt)*32 + lane] * 2 floats -> one b64 load per tile.
__global__ __launch_bounds__(256)
void chamfer_prep_ref(const float* __restrict__ ref,
                      float2* __restrict__ packed,
                      int N, int ntiles)
{
    const int g    = blockIdx.x * blockDim.x + threadIdx.x;
    const int lane = g & 31;
    const int tile = g >> 5;               // b*ntiles + nt
    const int b    = tile / ntiles;
    const int nt   = tile % ntiles;
    const int n    = nt * 16 + (lane & 15);

    const float* rp = ref + ((size_t)b * N + n) * 3;
    const float rx = rp[0], ry = rp[1], rz = rp[2];
    const bool hi = lane >= 16;

    float2 o;
    o.x = hi ? (-2.0f * rz) : (-2.0f * rx);
    o.y = hi ? (rx * rx + ry * ry + rz * rz) : (-2.0f * ry);
    packed[g] = o;
}

// One-sided chamfer with V_WMMA_F32_16X16X4_F32, C = inline 0.
// min_n P[m,n] = xx_m + min_n (yy_n - 2 q_m . r_n); xx folded into final sum.
// Ref tiles processed in pairs so accumulator updates fuse to v_min3_num_f32.
template <bool PREPPED>
__global__ __launch_bounds__(BLOCK_THREADS)
void chamfer_one_side_wmma(const float* __restrict__ query,
                           const float* __restrict__ refOrPacked,
                           float* __restrict__ out,
                           int N, int groupsPerBatch)
{
    const int lane  = threadIdx.x & 31;
    const int laneN = lane & 15;
    const bool hi   = lane >= 16;
    const int ntiles = N >> 4;

    const int gwave = blockIdx.x * WAVES_PER_BLOCK + (threadIdx.x >> 5);
    const int b     = gwave / groupsPerBatch;
    const int group = gwave % groupsPerBatch;

    const float* __restrict__ qb = query + (size_t)b * N * 3;

    // ---- Preamble: A matrices (K3 column = 1.0) + per-tile xx sums ----
    v2f   a[MT];
    float xxsum[MT];
    v8f   minv[MT];
#pragma unroll
    for (int mt = 0; mt < MT; ++mt) {
        const int m = (group * MT + mt) * 16 + laneN;
        const float qx = qb[3 * m + 0];
        const float qy = qb[3 * m + 1];
        const float qz = qb[3 * m + 2];
        a[mt].x = hi ? qz : qx;          // K0 / K2
        a[mt].y = hi ? 1.0f : qy;        // K1 / K3(=1.0 -> picks up yy_n)

        float xx = qx * qx + qy * qy + qz * qz;
#pragma unroll
        for (int off = 1; off < 16; off <<= 1)
            xx += __shfl_xor(xx, off, 32);   // sum over the 16-row tile
        xxsum[mt] = xx;

#pragma unroll
        for (int v = 0; v < 8; ++v) minv[mt][v] = 3.0e38f;
    }

    const v8f czero = {};   // folds to WMMA inline-0 SRC2

    // ---- Stream all ref tiles, two per step ----
    if (PREPPED) {
        const float2* __restrict__ p =
            (const float2*)refOrPacked + (size_t)b * ntiles * 32 + lane;
#pragma unroll 2
        for (int nt = 0; nt < ntiles; nt += 2) {
            const float2 bp0 = p[(size_t)nt * 32];        // one b64 each,
            const float2 bp1 = p[(size_t)(nt + 1) * 32];  // zero ALU prep
            v2f bm0; bm0.x = bp0.x; bm0.y = bp0.y;
            v2f bm1; bm1.x = bp1.x; bm1.y = bp1.y;
#pragma unroll
            for (int mt = 0; mt < MT; ++mt) {
                v8f d0 = __builtin_amdgcn_wmma_f32_16x16x4_f32(
                    false, a[mt], false, bm0, (short)0, czero, false, false);
                v8f d1 = __builtin_amdgcn_wmma_f32_16x16x4_f32(
                    false, a[mt], false, bm1, (short)0, czero, false, false);
#pragma unroll
                for (int v = 0; v < 8; ++v)
                    minv[mt][v] = fmin3_raw(minv[mt][v], d0[v], d1[v]);
            }
        }
    } else {
        for (int nt = 0; nt < ntiles; nt += 2) {
            v2f bm[2];
#pragma unroll
            for (int t = 0; t < 2; ++t) {
                const int n = (nt + t) * 16 + laneN;
                const float* rp = refOrPacked + ((size_t)b * N + n) * 3;
                const float rx = rp[0], ry = rp[1], rz = rp[2];
                bm[t].x = hi ? (-2.0f * rz) : (-2.0f * rx);
                bm[t].y = hi ? (rx * rx + ry * ry + rz * rz) : (-2.0f * ry);
            }
#pragma unroll
            for (int mt = 0; mt < MT; ++mt) {
                v8f d0 = __builtin_amdgcn_wmma_f32_16x16x4_f32(
                    false, a[mt], false, bm[0], (short)0, czero, false, false);
                v8f d1 = __builtin_amdgcn_wmma_f32_16x16x4_f32(
                    false, a[mt], false, bm[1], (short)0, czero, false, false);
#pragma unroll
                for (int v = 0; v < 8; ++v)
                    minv[mt][v] = fmin3_raw(minv[mt][v], d0[v], d1[v]);
            }
        }
    }

    // ---- Row-min reductions + wave contribution ----
    float s = 0.0f;
#pragma unroll
    for (int mt = 0; mt < MT; ++mt) {
#pragma unroll
        for (int v = 0; v < 8; ++v) {
#pragma unroll
            for (int off = 1; off < 16; off <<= 1)
                minv[mt][v] =
                    fmin_raw(minv[mt][v], __shfl_xor(minv[mt][v], off, 32));
        }
        float smt = 0.0f;
#pragma unroll
        for (int v = 0; v < 8; ++v) smt += minv[mt][v];
        if (!hi) smt += xxsum[mt];   // add sum_m xx_m exactly once per tile
        s += smt;
    }
    // lane 0 carries rows 0-7 (+xx sums), lane 16 carries rows 8-15.
    if (laneN == 0) atomicAdd(out, s);
}

extern "C" void kernel_launch(void* const* d_in, const int* in_sizes, int n_in,
                              void* d_out, int out_size, void* d_ws, size_t ws_size,
                              hipStream_t stream) {
    const float* preds = (const float*)d_in[0];
    const float* gts   = (const float*)d_in[1];
    float* out = (float*)d_out;

    const int N = 8192;
    const int total = in_sizes[0];          // B * N * 3
    const int B = total / (N * 3);
    const int ntiles = N / 16;
    const int groupsPerBatch = ntiles / MT;
    const int wavesTotal = B * groupsPerBatch;   // full waves only
    const int blocks = (wavesTotal + WAVES_PER_BLOCK - 1) / WAVES_PER_BLOCK;

    chamfer_zero_out<<<1, 1, 0, stream>>>(out);

    const size_t packedPerSide = (size_t)B * ntiles * 32;   // float2 records
    const size_t neededBytes = 2 * packedPerSide * sizeof(float2);

    if (ws_size >= neededBytes) {
        float2* packedGts   = (float2*)d_ws;
        float2* packedPreds = (float2*)d_ws + packedPerSide;

        const int prepThreads = B * ntiles * 32;
        const int prepBlocks  = (prepThreads + 255) / 256;
        chamfer_prep_ref<<<prepBlocks, 256, 0, stream>>>(gts,   packedGts,   N, ntiles);
        chamfer_prep_ref<<<prepBlocks, 256, 0, stream>>>(preds, packedPreds, N, ntiles);

        // loss_1: every pred point -> nearest gt point
        chamfer_one_side_wmma<true><<<blocks, BLOCK_THREADS, 0, stream>>>(
            preds, (const float*)packedGts, out, N, groupsPerBatch);
        // loss_2: every gt point -> nearest pred point
        chamfer_one_side_wmma<true><<<blocks, BLOCK_THREADS, 0, stream>>>(
            gts, (const float*)packedPreds, out, N, groupsPerBatch);
    } else {
        chamfer_one_side_wmma<false><<<blocks, BLOCK_THREADS, 0, stream>>>(
            preds, gts, out, N, groupsPerBatch);
        chamfer_one_side_wmma<false><<<blocks, BLOCK_THREADS, 0, stream>>>(
            gts, preds, out, N, groupsPerBatch);
    }
}